// Model_67284957659321
// MI455X (gfx1250) — compile-verified
//
#include <hip/hip_runtime.h>
#include <hip/hip_bf16.h>
#include <math.h>

#define Nn   8192
#define TOTv 262144
#define HIDv 18
#define NB   256   // blocks in first-stage final reduction

typedef float v2f __attribute__((ext_vector_type(2)));
typedef float v8f __attribute__((ext_vector_type(8)));

// ---- order-preserving float<->u32 mapping for integer atomic min/max ----
__device__ __forceinline__ unsigned f2ord(float f) {
    unsigned u = __float_as_uint(f);
    return (u & 0x80000000u) ? ~u : (u | 0x80000000u);
}
__device__ __forceinline__ float ord2f(unsigned u) {
    return (u & 0x80000000u) ? __uint_as_float(u & 0x7fffffffu)
                             : __uint_as_float(~u);
}

// ---------------- init: counters + min/max cells ----------------
__global__ void k_init(unsigned* cntA, unsigned* cntE, unsigned* mm) {
    int t = blockIdx.x * blockDim.x + threadIdx.x;
    if (t < Nn) { cntA[t] = 0u; cntE[t] = 0u; }
    if (t == 0) { mm[0] = 0xFFFFFFFFu; mm[1] = 0u; }  // min-init, max-init (ordered space)
}

// ---------------- graf0(AT): per-column count of AT>0, i!=j ----------------
// grid (8, 256), block 256: each thread owns 4 columns (float4), 32 rows per block.
__global__ void __launch_bounds__(256) k_colcountA(const float* __restrict__ AT,
                                                   unsigned* __restrict__ cntA) {
    int j4 = blockIdx.x * blockDim.x + threadIdx.x;   // 0..2047
    int j  = j4 * 4;
    int i0 = blockIdx.y * 32;
    unsigned c0 = 0, c1 = 0, c2 = 0, c3 = 0;
    for (int i = i0; i < i0 + 32; ++i) {
        float4 v = *(const float4*)(AT + (size_t)i * Nn + j);
        c0 += (v.x > 0.f && i != j);
        c1 += (v.y > 0.f && i != j + 1);
        c2 += (v.z > 0.f && i != j + 2);
        c3 += (v.w > 0.f && i != j + 3);
    }
    atomicAdd(&cntA[j], c0);     atomicAdd(&cntA[j + 1], c1);
    atomicAdd(&cntA[j + 2], c2); atomicAdd(&cntA[j + 3], c3);
}

// ---------------- normalize centered rows of ET: xn[r] = (ET[r]-mean)/||.|| ----------------
// one wave32 per row (FE == 32 == warpSize)
__global__ void __launch_bounds__(256) k_prep_xn(const float* __restrict__ ET,
                                                 float* __restrict__ xn) {
    int tid = threadIdx.x;
    int r = blockIdx.x * 8 + (tid >> 5);
    int k = tid & 31;
    float v = ET[r * 32 + k];
    float s = v;
    for (int d = 16; d > 0; d >>= 1) s += __shfl_xor(s, d, 32);
    float xm = v - s * (1.0f / 32.0f);
    float n2 = xm * xm;
    for (int d = 16; d > 0; d >>= 1) n2 += __shfl_xor(n2, d, 32);
    xn[r * 32 + k] = xm * rsqrtf(n2);
}

// ---------------- one 16x16 correlation tile via 8x V_WMMA_F32_16X16X4_F32 ----------------
__device__ __forceinline__ v8f corr_tile(const float* __restrict__ xn, int tm, int tn) {
    int lane = threadIdx.x;            // 0..31
    int m    = lane & 15;
    int h2   = (lane >> 4) * 2;        // 0 or 2 : K-pair selector per §7.12.2
    const float* arow = xn + (size_t)(tm * 16 + m) * 32 + h2;
    const float* brow = xn + (size_t)(tn * 16 + m) * 32 + h2;
    v8f acc = {};
#pragma unroll
    for (int kk = 0; kk < 8; ++kk) {
        v2f a = *(const v2f*)(arow + kk * 4);
        v2f b = *(const v2f*)(brow + kk * 4);
        acc = __builtin_amdgcn_wmma_f32_16x16x4_f32(false, a, false, b,
                                                    (short)0, acc, false, false);
    }
    return acc;
}

// ---------------- pass 1: global min/max of clip(corr) over lower triangle ----------------
// block (32,8): one wave per tile; grid (512, 64); wave-uniform skip keeps EXEC all-1s.
__global__ void __launch_bounds__(256) k_corr_minmax(const float* __restrict__ xn,
                                                     unsigned* __restrict__ mm) {
    int tn = blockIdx.x;
    int tm = blockIdx.y * 8 + threadIdx.y;
    if (tm < tn) return;
    v8f c = corr_tile(xn, tm, tn);
    float vmin = 1e30f, vmax = -1e30f;
#pragma unroll
    for (int r = 0; r < 8; ++r) {
        float v = fminf(1.0f, fmaxf(-1.0f, c[r]));
        vmin = fminf(vmin, v);
        vmax = fmaxf(vmax, v);
    }
    for (int d = 16; d > 0; d >>= 1) {
        vmin = fminf(vmin, __shfl_xor(vmin, d, 32));
        vmax = fmaxf(vmax, __shfl_xor(vmax, d, 32));
    }
    if (threadIdx.x == 0) {
        atomicMin(&mm[0], f2ord(vmin));
        atomicMax(&mm[1], f2ord(vmax));
    }
}

__global__ void k_mid(const unsigned* __restrict__ mm, float* __restrict__ mid) {
    float lo = ord2f(mm[0]), hi = ord2f(mm[1]);
    mid[0] = lo + (hi - lo) * 0.5f;
}

// ---------------- pass 2: per-column count of (c>=mid && i>j) ----------------
__global__ void __launch_bounds__(256) k_corr_count(const float* __restrict__ xn,
                                                    const float* __restrict__ midp,
                                                    unsigned* __restrict__ cntE) {
    int tn = blockIdx.x;
    int tm = blockIdx.y * 8 + threadIdx.y;
    if (tm < tn) return;
    float mid = midp[0];
    v8f c = corr_tile(xn, tm, tn);
    int lane  = threadIdx.x;
    int j     = tn * 16 + (lane & 15);
    int ibase = tm * 16 + ((lane >> 4) << 3);   // M = r (+8 for lanes 16..31)
    unsigned cnt = 0;
#pragma unroll
    for (int r = 0; r < 8; ++r) {
        float v = fminf(1.0f, fmaxf(-1.0f, c[r]));
        int i = ibase + r;
        cnt += (v >= mid && i > j) ? 1u : 0u;
    }
    cnt += __shfl_down(cnt, 16, 32);            // lanes L and L+16 share a column
    if (lane < 16) atomicAdd(&cntE[j], cnt);
}

// ---------------- fused stage: z = tanh(concat(H1, H2, E1, E2)) ----------------
__device__ __forceinline__ float dot32(const float* __restrict__ a,
                                       const float* __restrict__ w, int stride) {
    float s = 0.f;
#pragma unroll
    for (int k = 0; k < 32; ++k) s += a[k] * w[k * stride];
    return s;
}

__global__ void __launch_bounds__(256) k_stage(
    const float* __restrict__ XT, const float* __restrict__ ET, const float* __restrict__ YT,
    const float* __restrict__ W1, const float* __restrict__ W3,
    const float* __restrict__ W5, const float* __restrict__ W7,
    const float* __restrict__ b1, const float* __restrict__ b2,
    const float* __restrict__ b3, const float* __restrict__ b4,
    const float* __restrict__ L1w, const float* __restrict__ L1b,
    const float* __restrict__ L2w, const float* __restrict__ L2b,
    const float* __restrict__ L3w, const float* __restrict__ L3b,
    const float* __restrict__ L4w, const float* __restrict__ L4b,
    const unsigned* __restrict__ cntA, const unsigned* __restrict__ cntE,
    float* __restrict__ z) {
    int t = blockIdx.x * blockDim.x + threadIdx.x;
    if (t >= TOTv) return;
    float val;
    if (t < 65536) {                       // H1: [8192,8]
        int r = t >> 3, o = t & 7;
        float d = rsqrtf((float)(cntA[r] + 1u));
        val = d * dot32(XT + r * 32, W1 + o, 8) + b1[r];
    } else if (t < 131072) {               // H2: [16384,4]
        int u = t - 65536; int r = u >> 2, k = u & 3;
        float xy = (r < Nn) ? (dot32(XT + r * 32, L1w, 1) + L1b[0])
                            : (dot32(YT + (r - Nn) * 32, L2w, 1) + L2b[0]);
        val = xy * W5[k] + b3[r];
    } else if (t < 196608) {               // E1: [8192,8]
        int u = t - 131072; int r = u >> 3, o = u & 7;
        float d = rsqrtf((float)(cntE[r] + 1u));
        val = d * dot32(ET + r * 32, W3 + o, 8) + b2[r];
    } else {                               // E2: [16384,4]
        int u = t - 196608; int r = u >> 2, k = u & 3;
        float ey = (r < Nn) ? (dot32(ET + r * 32, L3w, 1) + L3b[0])
                            : (dot32(YT + (r - Nn) * 32, L4w, 1) + L4b[0]);
        val = ey * W7[k] + b4[r];
    }
    z[t] = tanhf(val);
}

// ---------------- stage-1 of h = z @ ler1_w.T : deterministic tree reduce ----------------
__global__ void __launch_bounds__(256) k_reduce1(const float* __restrict__ z,
                                                 const float* __restrict__ W,
                                                 float* __restrict__ partial) {
    __shared__ float sm[256];
    int tid = threadIdx.x;
    float acc[HIDv];
#pragma unroll
    for (int h = 0; h < HIDv; ++h) acc[h] = 0.f;
    for (int t = blockIdx.x * 256 + tid; t < TOTv; t += NB * 256) {
        float zt = z[t];
#pragma unroll
        for (int h = 0; h < HIDv; ++h) acc[h] += zt * W[(size_t)h * TOTv + t];
    }
    for (int h = 0; h < HIDv; ++h) {
        sm[tid] = acc[h];
        __syncthreads();
        for (int s = 128; s > 0; s >>= 1) {
            if (tid < s) sm[tid] += sm[tid + s];
            __syncthreads();
        }
        if (tid == 0) partial[blockIdx.x * HIDv + h] = sm[0];
        __syncthreads();
    }
}

// ---------------- final: sum partials (fixed order), bias, dot ler2 ----------------
__global__ void k_final(const float* __restrict__ partial, const float* __restrict__ l1b,
                        const float* __restrict__ l2w, const float* __restrict__ l2b,
                        float* __restrict__ out) {
    __shared__ float hbuf[HIDv];
    int tid = threadIdx.x;
    if (tid < HIDv) {
        float s = 0.f;
        for (int b = 0; b < NB; ++b) s += partial[b * HIDv + tid];
        hbuf[tid] = s + l1b[tid];
    }
    __syncthreads();
    if (tid == 0) {
        float o = l2b[0];
        for (int h = 0; h < HIDv; ++h) o += hbuf[h] * l2w[h];
        out[0] = o;
    }
}

extern "C" void kernel_launch(void* const* d_in, const int* in_sizes, int n_in,
                              void* d_out, int out_size, void* d_ws, size_t ws_size,
                              hipStream_t stream) {
    const float* AT  = (const float*)d_in[0];
    const float* XT  = (const float*)d_in[1];
    const float* ET  = (const float*)d_in[2];
    const float* YT  = (const float*)d_in[3];
    const float* W1  = (const float*)d_in[4];
    const float* W3  = (const float*)d_in[5];
    const float* W5  = (const float*)d_in[6];
    const float* W7  = (const float*)d_in[7];
    const float* b1  = (const float*)d_in[8];
    const float* b2  = (const float*)d_in[9];
    const float* b3  = (const float*)d_in[10];
    const float* b4  = (const float*)d_in[11];
    const float* L1w = (const float*)d_in[12];
    const float* L1b = (const float*)d_in[13];
    const float* L2w = (const float*)d_in[14];
    const float* L2b = (const float*)d_in[15];
    const float* L3w = (const float*)d_in[16];
    const float* L3b = (const float*)d_in[17];
    const float* L4w = (const float*)d_in[18];
    const float* L4b = (const float*)d_in[19];
    const float* l1w = (const float*)d_in[20];
    const float* l1b = (const float*)d_in[21];
    const float* l2w = (const float*)d_in[22];
    const float* l2b = (const float*)d_in[23];
    float* out = (float*)d_out;

    // workspace layout
    char* ws = (char*)d_ws;
    float*    xn      = (float*)(ws);                       // 8192*32 f32 = 1 MB
    unsigned* cntA    = (unsigned*)(ws + 1048576);          // 32 KB
    unsigned* cntE    = (unsigned*)(ws + 1048576 + 32768);  // 32 KB
    unsigned* mm      = (unsigned*)(ws + 1048576 + 65536);  // 2 u32
    float*    mid     = (float*)(ws + 1048576 + 65536 + 8); // 1 f32
    float*    z       = (float*)(ws + 1048576 + 65536 + 256);       // 1 MB
    float*    partial = (float*)(ws + 1048576 + 65536 + 256 + 1048576); // 256*18 f32

    k_init<<<dim3(32), dim3(256), 0, stream>>>(cntA, cntE, mm);
    k_colcountA<<<dim3(8, 256), dim3(256), 0, stream>>>(AT, cntA);
    k_prep_xn<<<dim3(1024), dim3(256), 0, stream>>>(ET, xn);
    k_corr_minmax<<<dim3(512, 64), dim3(32, 8), 0, stream>>>(xn, mm);
    k_mid<<<dim3(1), dim3(1), 0, stream>>>(mm, mid);
    k_corr_count<<<dim3(512, 64), dim3(32, 8), 0, stream>>>(xn, mid, cntE);
    k_stage<<<dim3(1024), dim3(256), 0, stream>>>(XT, ET, YT, W1, W3, W5, W7,
                                                  b1, b2, b3, b4,
                                                  L1w, L1b, L2w, L2b, L3w, L3b, L4w, L4b,
                                                  cntA, cntE, z);
    k_reduce1<<<dim3(NB), dim3(256), 0, stream>>>(z, l1w, partial);
    k_final<<<dim3(1), dim3(32), 0, stream>>>(partial, l1b, l2w, l2b, out);
    (void)in_sizes; (void)n_in; (void)out_size; (void)ws_size;
}